// MyVanillaRNN_56633438765601
// MI455X (gfx1250) — compile-verified
//
#include <hip/hip_runtime.h>
#include <math.h>

typedef float v2f __attribute__((ext_vector_type(2)));
typedef float v4f __attribute__((ext_vector_type(4)));
typedef float v8f __attribute__((ext_vector_type(8)));

#define RNN_B 64
#define RNN_T 512
#define RNN_I 512
#define RNN_H 1024

__device__ __forceinline__ v8f wmma_f32(v2f a, v2f b, v8f c) {
  // V_WMMA_F32_16X16X4_F32 : D(16x16 f32) = A(16x4) * B(4x16) + C
  return __builtin_amdgcn_wmma_f32_16x16x4_f32(false, a, false, b, (short)0, c,
                                               false, false);
}

// ---------------------------------------------------------------------------
// Kernel 1: x_proj = x @ Wx^T + bx  written directly into d_out (pre-act buf).
// M = B*T = 32768, N = H = 1024, K = I = 512.
// Each wave computes a 32x32 output block (2x2 WMMA tiles, 4 accumulators).
// A layout (16x4 f32): lane l<16 holds row (l), K = k0+{0,1}; lanes 16..31
// hold K = k0+{2,3}  -> one b64 load per lane per step.
// B layout (4x16): VGPR j holds K = k0+j (lanes 0-15) / K = k0+j+2 (16-31),
// N = n0 + (l&15); B[k][n] = Wx[n][k] -> one b64 load per lane per step.
// ---------------------------------------------------------------------------
__global__ __launch_bounds__(256) void xproj_kernel(
    const float* __restrict__ x, const float* __restrict__ Wx,
    const float* __restrict__ bx, float* __restrict__ out) {
  const int wave = threadIdx.x >> 5;
  const int lane = threadIdx.x & 31;
  const int lo = lane & 15;
  const int hi = lane >> 4;

  const int wid = blockIdx.x * 8 + wave;  // 32768 waves total
  const int mt = wid >> 5;                // 0..1023  (32-row blocks)
  const int nt = wid & 31;                // 0..31    (32-col blocks)
  const int r0 = mt * 32;
  const int n0 = nt * 32;

  const float* a0p = x + (size_t)(r0 + lo) * RNN_I + 2 * hi;
  const float* a1p = a0p + (size_t)16 * RNN_I;
  const float* b0p = Wx + (size_t)(n0 + lo) * RNN_I + 2 * hi;
  const float* b1p = b0p + (size_t)16 * RNN_I;

  v8f acc00 = {}, acc01 = {}, acc10 = {}, acc11 = {};
#pragma unroll 4
  for (int k = 0; k < RNN_I; k += 4) {
    v2f a0 = *(const v2f*)(a0p + k);
    v2f a1 = *(const v2f*)(a1p + k);
    v2f b0 = *(const v2f*)(b0p + k);
    v2f b1 = *(const v2f*)(b1p + k);
    acc00 = wmma_f32(a0, b0, acc00);
    acc01 = wmma_f32(a0, b1, acc01);
    acc10 = wmma_f32(a1, b0, acc10);
    acc11 = wmma_f32(a1, b1, acc11);
  }

  const float bias0 = bx[n0 + lo];
  const float bias1 = bx[n0 + 16 + lo];
#pragma unroll
  for (int i = 0; i < 8; ++i) {
    const int row0 = r0 + i + 8 * hi;  // C layout: VGPR i -> M=i / M=i+8
    const int row1 = row0 + 16;
    float* o0 = out + (size_t)row0 * RNN_H + n0 + lo;
    float* o1 = out + (size_t)row1 * RNN_H + n0 + lo;
    o0[0] = acc00[i] + bias0;
    o0[16] = acc01[i] + bias1;
    o1[0] = acc10[i] + bias0;
    o1[16] = acc11[i] + bias1;
  }
}

// ---------------------------------------------------------------------------
// Persistent scan kernel: 64 blocks x 128 threads (4 waves), all co-resident.
// Block nb owns output columns n0..n0+15; wave m owns batch rows m*16..m*16+15.
// Wh rows n0..n0+15 (64 KB) are preloaded into LDS once (XOR-swizzled to be
// bank-conflict-free for the per-lane b64 reads) and reused for all 512 steps.
// Per step:  acc = h_t @ Wh_tile^T  (256 x wmma f32 16x16x4, B from LDS),
//            pre = xp(t) + acc + bh  (read-modify-write d_out),
//            h_{t+1} = tanh(pre)  -> ping-pong buffer; grid barrier.
// ---------------------------------------------------------------------------
#define SWIZ(row, k) ((row)*RNN_H + ((k) ^ ((row) << 2)))

__device__ __forceinline__ void grid_sync(unsigned* cnt, unsigned target) {
  __builtin_amdgcn_fence(__ATOMIC_RELEASE, "agent");
  __syncthreads();
  if (threadIdx.x == 0) {
    __hip_atomic_fetch_add(cnt, 1u, __ATOMIC_RELAXED, __HIP_MEMORY_SCOPE_AGENT);
    while (__hip_atomic_load(cnt, __ATOMIC_RELAXED, __HIP_MEMORY_SCOPE_AGENT) <
           target) {
      __builtin_amdgcn_s_sleep(1);
    }
  }
  __syncthreads();
  __builtin_amdgcn_fence(__ATOMIC_ACQUIRE, "agent");
}

__global__ __launch_bounds__(128, 1) void rnn_scan_kernel(
    const float* __restrict__ Wh, const float* __restrict__ bh,
    float* __restrict__ out,      // [B,T,H], holds x_proj on entry
    float* __restrict__ hfin,     // [B,H] final hidden state
    float* __restrict__ h0buf, float* __restrict__ h1buf,
    unsigned* __restrict__ cnt) {
  __shared__ float whs[16 * RNN_H];  // 64 KB

  const int n0 = blockIdx.x * 16;
  const int wave = threadIdx.x >> 5;  // m-tile 0..3
  const int lane = threadIdx.x & 31;
  const int lo = lane & 15;
  const int hi = lane >> 4;

  // Preload Wh rows n0..n0+15 into LDS (swizzled, float4).
  for (int idx = threadIdx.x; idx < 16 * (RNN_H / 4); idx += 128) {
    const int row = idx >> 8;            // 256 float4 per row
    const int c4 = (idx & 255) << 2;
    v4f v = *(const v4f*)(Wh + (size_t)(n0 + row) * RNN_H + c4);
    *(v4f*)(&whs[SWIZ(row, c4)]) = v;
  }
  __syncthreads();

  const int arow = (wave * 16 + lo) * RNN_H + 2 * hi;  // A offset in h buffer
  const int xorv = lo << 2;                            // LDS bank swizzle
  const float* bbase = whs + lo * RNN_H + 2 * hi;
  const float bias = bh[n0 + lo];
  const int n = n0 + lo;

  for (int t = 0; t < RNN_T; ++t) {
    const float* hc = (t & 1) ? h1buf : h0buf;
    float* hn = (t & 1) ? h0buf : h1buf;
    const float* ap = hc + arow;

    v8f acc = {};
#pragma unroll 8
    for (int k = 0; k < RNN_H; k += 4) {
      v2f a = *(const v2f*)(ap + k);                  // global b64 (L2-hot)
      v2f b = *(const v2f*)(bbase + (k ^ xorv));      // ds_load_b64
      acc = wmma_f32(a, b, acc);
    }

#pragma unroll
    for (int i = 0; i < 8; ++i) {
      const int row = wave * 16 + i + 8 * hi;
      const size_t oidx =
          (size_t)row * (RNN_T * RNN_H) + (size_t)t * RNN_H + n;
      const float pre = out[oidx] + acc[i] + bias;
      out[oidx] = pre;                 // stored output is the pre-activation
      const float h = tanhf(pre);
      hn[row * RNN_H + n] = h;
      if (t == RNN_T - 1) hfin[row * RNN_H + n] = h;
    }
    grid_sync(cnt, (unsigned)(64 * (t + 1)));
  }
}

// ---------------------------------------------------------------------------
extern "C" void kernel_launch(void* const* d_in, const int* in_sizes, int n_in,
                              void* d_out, int out_size, void* d_ws,
                              size_t ws_size, hipStream_t stream) {
  (void)in_sizes; (void)n_in; (void)out_size; (void)ws_size;
  const float* x = (const float*)d_in[0];
  const float* h_0 = (const float*)d_in[1];
  const float* Wx = (const float*)d_in[2];
  const float* bx = (const float*)d_in[3];
  const float* Wh = (const float*)d_in[4];
  const float* bh = (const float*)d_in[5];

  float* out = (float*)d_out;
  float* hfin = out + (size_t)RNN_B * RNN_T * RNN_H;

  unsigned char* ws = (unsigned char*)d_ws;
  unsigned* cnt = (unsigned*)ws;                       // barrier counter
  float* h0buf = (float*)(ws + 1024);                  // [B,H]
  float* h1buf = (float*)(ws + 1024 + (size_t)RNN_B * RNN_H * sizeof(float));

  hipMemsetAsync(cnt, 0, 1024, stream);
  hipMemcpyAsync(h0buf, h_0, (size_t)RNN_B * RNN_H * sizeof(float),
                 hipMemcpyDeviceToDevice, stream);

  xproj_kernel<<<4096, 256, 0, stream>>>(x, Wx, bx, out);
  rnn_scan_kernel<<<64, 128, 0, stream>>>(Wh, bh, out, hfin, h0buf, h1buf,
                                          cnt);
}